// Sampler_64003602645551
// MI455X (gfx1250) — compile-verified
//
#include <hip/hip_runtime.h>
#include <stdint.h>

#define BLOCK   256
#define NBIN1   2048
#define NBIN2   2048
#define NBIN3   1024
#define HASHN   4096
#define HASHM   (HASHN - 1)
#define HEMPTY  0xFFFFFFFFu
#define CHUNK   1024          /* dwords per streamed chunk = BLOCK*4 */
#define TOPK_LP 5

struct Smem {
  float stage[2][CHUNK];               // MUST be first: async-LDS staging (8 KB)
  unsigned hash[HASHN];                // token hash: [16:0]=tok [24:17]=count [25]=prompt (16 KB)
  unsigned long long hist[NBIN1];      // radix histograms: counts or fixed-point exp mass (16 KB)
  unsigned long long maxcell;          // packed argmax reduction cell
  unsigned long long wincell;          // top-5 merge cell
  unsigned bc_sel;                     // broadcast: selected bin
  unsigned bc_r;                       // broadcast: residual rank
  unsigned long long bc_crem;          // broadcast: residual mass budget
};

// ---- float <-> totally-ordered uint (bijective) ----
__device__ __forceinline__ unsigned srt_u(float f) {
  unsigned u = __float_as_uint(f);
  return (u & 0x80000000u) ? ~u : (u | 0x80000000u);
}
__device__ __forceinline__ float unsrt_f(unsigned u) {
  unsigned s = (u & 0x80000000u) ? (u & 0x7FFFFFFFu) : ~u;
  return __uint_as_float(s);
}

// ---- LDS token hash ----
__device__ __forceinline__ unsigned hash0(unsigned tok) {
  return ((tok * 2654435761u) >> 20) & HASHM;
}
__device__ __forceinline__ unsigned hash_insert(unsigned* h, unsigned tok) {
  unsigned i = hash0(tok);
  for (;;) {
    unsigned cur = h[i];
    if ((cur & 0x1FFFFu) == tok && cur != HEMPTY) return i;
    if (cur == HEMPTY) {
      unsigned prev = atomicCAS(&h[i], HEMPTY, tok);
      if (prev == HEMPTY || (prev & 0x1FFFFu) == tok) return i;
    }
    i = (i + 1) & HASHM;
  }
}
__device__ __forceinline__ unsigned hash_lookup(const unsigned* h, unsigned tok) {
  unsigned i = hash0(tok);
  for (;;) {
    unsigned cur = h[i];
    if (cur == HEMPTY) return 0u;                 // miss (any hit has count or prompt bit set)
    if ((cur & 0x1FFFFu) == tok) return cur;
    i = (i + 1) & HASHM;
  }
}

// ---- penalties; explicit _rn ops so recompute paths are bit-identical ----
__device__ __forceinline__ float penalize(float x, unsigned info, float rep, float freq,
                                          float pres, float temp_eff) {
  float l = x;
  if (info) {                                      // occurred = prompt | count>0
    unsigned cnt = (info >> 17) & 0xFFu;
    l = (l > 0.0f) ? __fdiv_rn(l, rep) : __fmul_rn(l, rep);
    l = __fsub_rn(l, __fmul_rn(freq, (float)cnt));
    if (cnt) l = __fsub_rn(l, pres);
  }
  return __fdiv_rn(l, temp_eff);
}

// ---- async global->LDS primitives (CDNA5 ASYNCcnt path) ----
__device__ __forceinline__ void async_b128(unsigned lds_off, const void* gp) {
  uint64_t g = (uint64_t)(uintptr_t)gp;
  asm volatile("global_load_async_to_lds_b128 %0, %1, off" :: "v"(lds_off), "v"(g) : "memory");
}
__device__ __forceinline__ void wait_async0() { asm volatile("s_wait_asynccnt 0x0" ::: "memory"); }
__device__ __forceinline__ void wait_async1() { asm volatile("s_wait_asynccnt 0x1" ::: "memory"); }

// ---- per-wave double-buffered streamer over 32-bit words ----
template <typename F>
__device__ __forceinline__ void stream_row_u(const unsigned* __restrict__ row, Smem& sm,
                                             int V, F&& body) {
  const int tid = (int)threadIdx.x;
  const unsigned sbase = (unsigned)(uintptr_t)(&sm.stage[0][0]);   // LDS byte offset
  const int nch = V / CHUNK;
  async_b128(sbase + (unsigned)tid * 16u, row + (size_t)tid * 4);
  for (int c = 0; c < nch; ++c) {
    const int buf = c & 1;
    if (c + 1 < nch) {
      async_b128(sbase + (unsigned)((buf ^ 1) * (CHUNK * 4)) + (unsigned)tid * 16u,
                 row + (size_t)(c + 1) * CHUNK + (size_t)tid * 4);
      wait_async1();
    } else {
      wait_async0();
    }
    const uint4 q = *reinterpret_cast<const uint4*>(
        reinterpret_cast<const unsigned*>(&sm.stage[buf][0]) + tid * 4);
    const int v0 = c * CHUNK + tid * 4;
    body(v0 + 0, q.x); body(v0 + 1, q.y); body(v0 + 2, q.z); body(v0 + 3, q.w);
  }
}

__device__ __forceinline__ void zero_hist(Smem& sm) {
  for (int i = (int)threadIdx.x; i < NBIN1; i += BLOCK) sm.hist[i] = 0ull;
}

// thread-0 scans -------------------------------------------------------------
__device__ __forceinline__ void scan_count(Smem& sm, int nbins, unsigned target) {
  unsigned long long cum = 0; unsigned sel = 0, rr = 1;
  for (int b = nbins - 1; b >= 0; --b) {
    unsigned long long cb = sm.hist[b];
    if (cum + cb >= (unsigned long long)target) { sel = (unsigned)b; rr = target - (unsigned)cum; break; }
    cum += cb;
  }
  sm.bc_sel = sel; sm.bc_r = rr;
}
__device__ __forceinline__ void scan_mass(Smem& sm, int nbins, unsigned long long c) {
  unsigned long long A = 0, crem = 0; unsigned sel = 0;
  for (int b = nbins - 1; b >= 0; --b) {
    if (A > c) break;
    unsigned long long mb = sm.hist[b];
    if (mb) { sel = (unsigned)b; crem = c - A; }
    A += mb;
  }
  sm.bc_sel = sel; sm.bc_crem = crem;
}

template <bool CACHE>
__global__ void __launch_bounds__(BLOCK) sampler_kernel(
    const float* __restrict__ logits, const float* __restrict__ noise,
    const int* __restrict__ ptoks, const int* __restrict__ otoks,
    const float* __restrict__ presence, const float* __restrict__ freqp,
    const float* __restrict__ repp, const float* __restrict__ temperature,
    const int* __restrict__ topk, const float* __restrict__ topp,
    float* __restrict__ out, unsigned* __restrict__ ws,
    int V, int B, int PL, int OL) {
  __shared__ Smem sm;
  const int row = (int)blockIdx.x;
  const int tid = (int)threadIdx.x;
  const float* rowlog = logits + (size_t)row * V;
  const float* rownoi = noise  + (size_t)row * V;
  unsigned* wsrow = CACHE ? (ws + (size_t)row * V) : nullptr;

  for (int i = tid; i < HASHN; i += BLOCK) sm.hash[i] = HEMPTY;
  zero_hist(sm);
  if (tid == 0) sm.maxcell = 0ull;
  __syncthreads();

  // ---- scatter penalties into hash ----
  for (int i = tid; i < OL; i += BLOCK) {
    unsigned s = hash_insert(sm.hash, (unsigned)otoks[(size_t)row * OL + i]);
    atomicAdd(&sm.hash[s], 1u << 17);
  }
  for (int i = tid; i < PL; i += BLOCK) {
    unsigned s = hash_insert(sm.hash, (unsigned)ptoks[(size_t)row * PL + i]);
    atomicOr(&sm.hash[s], 1u << 25);
  }
  const float rep = repp[row], freq = freqp[row], pres = presence[row];
  const float temp = temperature[row];
  const bool greedy_row = temp < 1e-5f;
  const float temp_eff = greedy_row ? 1.0f : temp;
  int k = topk[row]; if (k < 1) k = 1; if (k > V) k = V;
  const float tp = topp[row];
  __syncthreads();

  // ---- P1: raw top-5, greedy argmax, level-1 count histogram, (opt) key cache ----
  unsigned long long t5[TOPK_LP] = {0, 0, 0, 0, 0};
  unsigned long long gmax = 0ull;
  {
    const unsigned sbase = (unsigned)(uintptr_t)(&sm.stage[0][0]);
    const unsigned* rl = reinterpret_cast<const unsigned*>(rowlog);
    const int nch = V / CHUNK;
    async_b128(sbase + (unsigned)tid * 16u, rl + (size_t)tid * 4);
    for (int c = 0; c < nch; ++c) {
      const int buf = c & 1;
      if (c + 1 < nch) {
        async_b128(sbase + (unsigned)((buf ^ 1) * (CHUNK * 4)) + (unsigned)tid * 16u,
                   rl + (size_t)(c + 1) * CHUNK + (size_t)tid * 4);
        wait_async1();
      } else {
        wait_async0();
      }
      const uint4 q = *reinterpret_cast<const uint4*>(
          reinterpret_cast<const unsigned*>(&sm.stage[buf][0]) + tid * 4);
      const unsigned qa[4] = {q.x, q.y, q.z, q.w};
      const int v0 = c * CHUNK + tid * 4;
      unsigned us[4];
#pragma unroll
      for (int j = 0; j < 4; ++j) {
        const int v = v0 + j;
        const float raw = __uint_as_float(qa[j]);
        unsigned long long rk =
            ((unsigned long long)srt_u(raw) << 32) | (unsigned)(~(unsigned)v);
        if (rk > t5[TOPK_LP - 1]) {
          t5[TOPK_LP - 1] = rk;
          for (int s = TOPK_LP - 1; s > 0 && t5[s] > t5[s - 1]; --s) {
            unsigned long long tmp = t5[s]; t5[s] = t5[s - 1]; t5[s - 1] = tmp;
          }
        }
        unsigned info = hash_lookup(sm.hash, (unsigned)v);
        float l = penalize(raw, info, rep, freq, pres, temp_eff);
        unsigned u = srt_u(l);
        us[j] = u;
        atomicAdd(&sm.hist[u >> 21], 1ull);
        unsigned long long gk = ((unsigned long long)u << 32) | (unsigned)(~(unsigned)v);
        if (gk > gmax) gmax = gk;
      }
      if constexpr (CACHE) {
        *reinterpret_cast<uint4*>(wsrow + v0) = make_uint4(us[0], us[1], us[2], us[3]);
      }
    }
    if constexpr (CACHE) {
      asm volatile("s_wait_storecnt 0x0" ::: "memory");  // keys visible in GL2 for async reads
      __threadfence();
    }
  }
  atomicMax(&sm.maxcell, gmax);
  __syncthreads();
  const unsigned long long G = sm.maxcell;
  const unsigned greedy_idx = ~(unsigned)(G & 0xFFFFFFFFull);
  const float M = unsrt_f((unsigned)(G >> 32));            // row max of penalized logits

  // ---- merge raw top-5 across threads (5 exclusion rounds) ----
  int p5 = 0;
  for (int r = 0; r < TOPK_LP; ++r) {
    if (tid == 0) sm.wincell = 0ull;
    __syncthreads();
    if (p5 < TOPK_LP) atomicMax(&sm.wincell, t5[p5]);
    __syncthreads();
    unsigned long long w = sm.wincell;
    if (p5 < TOPK_LP && t5[p5] == w) ++p5;
    if (tid == 0) {
      out[(size_t)B + (size_t)row * TOPK_LP + r] = unsrt_f((unsigned)(w >> 32));
      ((int*)out)[(size_t)B + (size_t)B * TOPK_LP + (size_t)row * TOPK_LP + r] =
          (int)(~(unsigned)(w & 0xFFFFFFFFull));
    }
    __syncthreads();
  }

  // source of penalized keys for passes 2..7
  const unsigned* usrc = CACHE ? (const unsigned*)wsrow : (const unsigned*)rowlog;
  auto key_of = [&](int v, unsigned w) -> unsigned {
    if constexpr (CACHE) {
      return w;                                    // cached sortable key
    } else {
      unsigned info = hash_lookup(sm.hash, (unsigned)v);
      return srt_u(penalize(__uint_as_float(w), info, rep, freq, pres, temp_eff));
    }
  };

  // ---- radix-select kth largest (top-k threshold), 3 levels ----
  if (tid == 0) scan_count(sm, NBIN1, (unsigned)k);
  __syncthreads();
  const unsigned b1 = sm.bc_sel; const unsigned r1 = sm.bc_r;
  zero_hist(sm); __syncthreads();

  stream_row_u(usrc, sm, V, [&](int v, unsigned w) {
    unsigned u = key_of(v, w);
    if ((u >> 21) == b1) atomicAdd(&sm.hist[(u >> 10) & 0x7FFu], 1ull);
  });
  __syncthreads();
  if (tid == 0) scan_count(sm, NBIN2, r1);
  __syncthreads();
  const unsigned b2 = sm.bc_sel; const unsigned r2 = sm.bc_r;
  zero_hist(sm); __syncthreads();

  stream_row_u(usrc, sm, V, [&](int v, unsigned w) {
    unsigned u = key_of(v, w);
    if ((u >> 10) == ((b1 << 11) | b2)) atomicAdd(&sm.hist[u & 0x3FFu], 1ull);
  });
  __syncthreads();
  if (tid == 0) scan_count(sm, NBIN3, r2);
  __syncthreads();
  const unsigned kv = (b1 << 21) | (b2 << 10) | sm.bc_sel;   // sortable bits of kth value
  zero_hist(sm); __syncthreads();

  // ---- top-p threshold: 3-level descent over fixed-point exp mass (2^43) ----
  const float ESC = 8796093022208.0f;   // 2^43; e<=1 after subtracting M
  stream_row_u(usrc, sm, V, [&](int v, unsigned w) {
    unsigned u = key_of(v, w);
    if (u >= kv) {
      float l = unsrt_f(u);
      unsigned long long m = (unsigned long long)(__fmul_rn(expf(__fsub_rn(l, M)), ESC));
      atomicAdd(&sm.hist[u >> 21], m);
    }
  });
  __syncthreads();
  if (tid == 0) {
    unsigned long long Z = 0;
    for (int b = 0; b < NBIN1; ++b) Z += sm.hist[b];
    scan_mass(sm, NBIN1, (unsigned long long)((double)tp * (double)Z));
  }
  __syncthreads();
  const unsigned e1 = sm.bc_sel; const unsigned long long c1 = sm.bc_crem;
  zero_hist(sm); __syncthreads();

  stream_row_u(usrc, sm, V, [&](int v, unsigned w) {
    unsigned u = key_of(v, w);
    if (u >= kv && (u >> 21) == e1) {
      float l = unsrt_f(u);
      unsigned long long m = (unsigned long long)(__fmul_rn(expf(__fsub_rn(l, M)), ESC));
      atomicAdd(&sm.hist[(u >> 10) & 0x7FFu], m);
    }
  });
  __syncthreads();
  if (tid == 0) scan_mass(sm, NBIN2, c1);
  __syncthreads();
  const unsigned e2 = sm.bc_sel; const unsigned long long c2 = sm.bc_crem;
  zero_hist(sm); __syncthreads();

  stream_row_u(usrc, sm, V, [&](int v, unsigned w) {
    unsigned u = key_of(v, w);
    if (u >= kv && (u >> 10) == ((e1 << 11) | e2)) {
      float l = unsrt_f(u);
      unsigned long long m = (unsigned long long)(__fmul_rn(expf(__fsub_rn(l, M)), ESC));
      atomicAdd(&sm.hist[u & 0x3FFu], m);
    }
  });
  __syncthreads();
  if (tid == 0) { scan_mass(sm, NBIN3, c2); sm.maxcell = 0ull; }
  __syncthreads();
  const unsigned tau = (e1 << 21) | (e2 << 10) | sm.bc_sel;  // nucleus cutoff (>= kv)

  // ---- P7: Gumbel argmax over kept set; dual async stream (keys + noise) ----
  {
    const unsigned sbase = (unsigned)(uintptr_t)(&sm.stage[0][0]);
    unsigned long long rmax = 0ull;
    const int nch = V / CHUNK;
    for (int c = 0; c < nch; ++c) {
      async_b128(sbase + (unsigned)tid * 16u,
                 usrc + (size_t)c * CHUNK + (size_t)tid * 4);
      async_b128(sbase + (unsigned)(CHUNK * 4) + (unsigned)tid * 16u,
                 rownoi + (size_t)c * CHUNK + (size_t)tid * 4);
      wait_async0();
      const uint4 qv = *reinterpret_cast<const uint4*>(
          reinterpret_cast<const unsigned*>(&sm.stage[0][0]) + tid * 4);
      const float4 nv = *reinterpret_cast<const float4*>(&sm.stage[1][tid * 4]);
      const unsigned qw[4] = {qv.x, qv.y, qv.z, qv.w};
      const float nw[4] = {nv.x, nv.y, nv.z, nv.w};
      const int v0 = c * CHUNK + tid * 4;
#pragma unroll
      for (int j = 0; j < 4; ++j) {
        const int v = v0 + j;
        unsigned u = key_of(v, qw[j]);
        if (u >= tau) {
          float l = unsrt_f(u);
          float gum = -logf(-logf(nw[j]));
          unsigned long long key =
              ((unsigned long long)srt_u(l + gum) << 32) | (unsigned)(~(unsigned)v);
          if (key > rmax) rmax = key;
        }
      }
    }
    atomicMax(&sm.maxcell, rmax);
  }
  __syncthreads();
  if (tid == 0) {
    unsigned rnd_idx = ~(unsigned)(sm.maxcell & 0xFFFFFFFFull);
    ((int*)out)[row] = greedy_row ? (int)greedy_idx : (int)rnd_idx;
  }
}

extern "C" void kernel_launch(void* const* d_in, const int* in_sizes, int n_in,
                              void* d_out, int out_size, void* d_ws, size_t ws_size,
                              hipStream_t stream) {
  (void)n_in; (void)out_size;
  const float* logits = (const float*)d_in[0];
  const float* noise  = (const float*)d_in[1];
  const int*   ptoks  = (const int*)d_in[2];
  const int*   otoks  = (const int*)d_in[3];
  const float* pres   = (const float*)d_in[4];
  const float* freq   = (const float*)d_in[5];
  const float* rep    = (const float*)d_in[6];
  const float* temp   = (const float*)d_in[7];
  const int*   topk   = (const int*)d_in[8];
  const float* topp   = (const float*)d_in[9];
  const int B  = in_sizes[4];
  const int V  = in_sizes[0] / B;   // 128000, multiple of CHUNK=1024
  const int PL = in_sizes[2] / B;
  const int OL = in_sizes[3] / B;
  const size_t need = (size_t)B * (size_t)V * sizeof(unsigned);
  if (d_ws != nullptr && ws_size >= need) {
    sampler_kernel<true><<<B, BLOCK, 0, stream>>>(logits, noise, ptoks, otoks, pres, freq,
                                                  rep, temp, topk, topp, (float*)d_out,
                                                  (unsigned*)d_ws, V, B, PL, OL);
  } else {
    sampler_kernel<false><<<B, BLOCK, 0, stream>>>(logits, noise, ptoks, otoks, pres, freq,
                                                   rep, temp, topk, topp, (float*)d_out,
                                                   nullptr, V, B, PL, OL);
  }
}